// Segmenter_23596550324744
// MI455X (gfx1250) — compile-verified
//
#include <hip/hip_runtime.h>
#include <stdint.h>
#include <math.h>

#define SEG  1024
#define HALF 512

// y[b,n] = x[b,n] * W(n mod 512),  W(r) = sqrt(win[r])^2 + sqrt(win[r+512])^2
// Exact collapse of the WOLA segment -> window -> overlap-add -> window chain
// with S=1024, hop=512 and edge-corrected pre/post windows: every sample's
// total squared-window weight is the same 512-periodic function, including at
// both edges (that is precisely what the pre/post edge correction ensures).
__global__ __launch_bounds__(256) void
Segmenter_23596550324744_kernel(const float* __restrict__ x,
                                const float* __restrict__ window,
                                float* __restrict__ y,
                                long long total4, long long total_n)
{
    __shared__ float s_win[SEG];   // raw window, filled by async global->LDS DMA
    __shared__ float s_w[HALF];    // periodic per-sample weight

    const int t = threadIdx.x;

    // --- gfx1250 async copy: 256 lanes x b128 = 4KB window into LDS ---------
    {
        uint32_t lds_off = (uint32_t)(uintptr_t)(&s_win[0]) + (uint32_t)(t * 16);
        uint32_t vbyte   = (uint32_t)(t * 16);
        asm volatile(
            "global_load_async_to_lds_b128 %0, %1, %2\n\t"
            "s_wait_asynccnt 0x0"
            :
            : "v"(lds_off), "v"(vbyte), "s"(window)
            : "memory");   // memory clobber = compiler barrier for s_win reads
    }
    __syncthreads();

    // --- build 512-entry weight table (sqrt-then-square mimics reference) ---
    {
        float a0 = s_win[t],        a1 = s_win[t + HALF];
        float b0 = s_win[t + 256],  b1 = s_win[t + 256 + HALF];
        float qa0 = sqrtf(a0), qa1 = sqrtf(a1);
        float qb0 = sqrtf(b0), qb1 = sqrtf(b1);
        s_w[t]       = qa0 * qa0 + qa1 * qa1;
        s_w[t + 256] = qb0 * qb0 + qb1 * qb1;
    }
    __syncthreads();

    // --- bandwidth-bound stream --------------------------------------------
    // blockDim = 256 => grid stride (in float4 units) is a multiple of 128
    // (one 512-float period), and blockIdx*256 is too, so the weight index
    // i & 127 == t & 127 is loop-invariant: load it ONCE into registers.
    const float4* __restrict__ x4 = (const float4*)x;
    float4* __restrict__       y4 = (float4*)y;
    const float4 wv = ((const float4*)s_w)[t & 127];

    long long i      = (long long)blockIdx.x * blockDim.x + t;
    long long stride = (long long)gridDim.x * blockDim.x;
    for (; i < total4; i += stride) {
        float4 v = x4[i];          // global_load_b128
        float4 o;
        o.x = v.x * wv.x;
        o.y = v.y * wv.y;
        o.z = v.z * wv.z;
        o.w = v.w * wv.w;
        y4[i] = o;                 // global_store_b128
    }

    // scalar tail (not hit for the given sizes, kept for robustness)
    for (long long j = total4 * 4 + (long long)blockIdx.x * blockDim.x + t;
         j < total_n; j += stride) {
        y[j] = x[j] * s_w[(int)(j & 511)];
    }
}

extern "C" void kernel_launch(void* const* d_in, const int* in_sizes, int n_in,
                              void* d_out, int out_size, void* d_ws, size_t ws_size,
                              hipStream_t stream) {
    const float* x      = (const float*)d_in[0];   // [16, 2097152] f32
    const float* window = (const float*)d_in[1];   // [1024] f32
    float*       y      = (float*)d_out;           // [16, 2097152] f32

    long long n      = (long long)in_sizes[0];
    long long total4 = n >> 2;

    const int block = 256;               // 8 wave32 per workgroup
    long long nb = (total4 + block - 1) / block;
    int grid = 8192;                     // 4 float4 per thread; grid-stride
    if (nb < grid) grid = (int)(nb > 0 ? nb : 1);

    Segmenter_23596550324744_kernel<<<grid, block, 0, stream>>>(
        x, window, y, total4, n);
}